// RPE_47364899340507
// MI455X (gfx1250) — compile-verified
//
#include <hip/hip_runtime.h>
#include <cstdint>

#define WDIM   48
#define NPTS   2304                    // 48*48
#define NN     (NPTS * NPTS)
#define RPE    33
#define HEADS  8
#define TBL_FLOATS (3 * RPE * HEADS)   // 792 floats = 3168 bytes = 198 x 16B

typedef float v4f __attribute__((ext_vector_type(4)));
typedef __attribute__((address_space(3))) float lds_float_t;

// ---------------------------------------------------------------------------
// Kernel 1: per-image depth min/max + normalization.  depth (2,48,48) -> z ws
// ---------------------------------------------------------------------------
__global__ __launch_bounds__(256) void rpe_zprep_kernel(
    const float* __restrict__ depth, float* __restrict__ z)
{
    const int b = blockIdx.x;
    const int t = threadIdx.x;
    const float* d = depth + (size_t)b * NPTS;

    float v[9];
    float mn = 1e30f, mx = -1e30f;
#pragma unroll
    for (int k = 0; k < 9; ++k) {
        v[k] = d[t + 256 * k];
        mn = fminf(mn, v[k]);
        mx = fmaxf(mx, v[k]);
    }

    __shared__ float smn[256];
    __shared__ float smx[256];
    smn[t] = mn; smx[t] = mx;
    __syncthreads();
#pragma unroll
    for (int s = 128; s > 0; s >>= 1) {
        if (t < s) {
            smn[t] = fminf(smn[t], smn[t + s]);
            smx[t] = fmaxf(smx[t], smx[t + s]);
        }
        __syncthreads();
    }
    const float zmin = smn[0];
    const float zden = smx[0] - smn[0] + 1e-8f;
#pragma unroll
    for (int k = 0; k < 9; ++k)
        z[(size_t)b * NPTS + t + 256 * k] = (v[k] - zmin) / zden;
}

// ---------------------------------------------------------------------------
// Kernel 2: out[b,h,i,j] = sum_axis table[q+16+axis*33, h]
// grid: (i = 2304, b = 2), 192 threads; thread handles 4 consecutive j
// (j0 % 4 == 0 and 48 % 4 == 0 -> the 4-group never crosses a row, so the
//  y-row gather is shared across the group). 192*4*3 = 2304 = N exactly.
// ---------------------------------------------------------------------------
__global__ __launch_bounds__(192) void rpe_main_kernel(
    const float* __restrict__ table,   // (99, 8)
    const float* __restrict__ z,       // (2, 2304) normalized
    float* __restrict__ out)           // (2, 8, 2304, 2304)
{
    __shared__ __align__(16) float tbl[TBL_FLOATS];

    const int t = threadIdx.x;
    const int i = blockIdx.x;
    const int b = blockIdx.y;

    // ---- async copy of the RPE table (3168 B) into LDS: 198 x b128 ----
    {
        lds_float_t* lp = (lds_float_t*)&tbl[0];
        const unsigned ldsBase = (unsigned)(uintptr_t)lp;
        for (int k = t; k < TBL_FLOATS / 4; k += 192) {
            unsigned ldsAddr = ldsBase + (unsigned)k * 16u;
            const float* gaddr = table + k * 4;
            asm volatile("global_load_async_to_lds_b128 %0, %1, off"
                         :: "v"(ldsAddr), "v"(gaddr) : "memory");
        }
    }
    asm volatile("s_wait_asynccnt 0" ::: "memory");
    __syncthreads();

    const v4f* t4 = reinterpret_cast<const v4f*>(tbl);

    // per-row-i scalars
    const int   ri   = i / WDIM;
    const int   ci   = i - ri * WDIM;
    const float K    = 16.0f / 47.0f;
    const float xi16 = (float)ci * K;      // x_i * 16
    const float yi16 = (float)ri * K;      // y_i * 16
    const float zi16 = z[(size_t)b * NPTS + i] * 16.0f;
    const v4f*  z4   = reinterpret_cast<const v4f*>(z + (size_t)b * NPTS);

    // per-head uniform output bases (kept in SGPRs; stores use saddr+voffset)
    const size_t ibase = (size_t)i * NPTS;
    float* oh0 = out + ((size_t)(b * HEADS) + 0) * NN + ibase;
    float* oh1 = out + ((size_t)(b * HEADS) + 1) * NN + ibase;
    float* oh2 = out + ((size_t)(b * HEADS) + 2) * NN + ibase;
    float* oh3 = out + ((size_t)(b * HEADS) + 3) * NN + ibase;
    float* oh4 = out + ((size_t)(b * HEADS) + 4) * NN + ibase;
    float* oh5 = out + ((size_t)(b * HEADS) + 5) * NN + ibase;
    float* oh6 = out + ((size_t)(b * HEADS) + 6) * NN + ibase;
    float* oh7 = out + ((size_t)(b * HEADS) + 7) * NN + ibase;

#pragma unroll 1
    for (int jt = 0; jt < 3; ++jt) {
        const int j0 = jt * 768 + 4 * t;         // multiple of 4
        const int rj = j0 / WDIM;
        const float cjf = (float)(j0 - rj * WDIM);

        // y quantization: shared by the 4-group (row constant)
        float fy = rintf(fmaf(-(float)rj, K, yi16));
        fy = fminf(fmaxf(fy, -16.0f), 16.0f);
        const int iy = (int)fy + 49;             // [33, 66)
        const v4f ya = t4[2 * iy], yb = t4[2 * iy + 1];

        const v4f zj4 = z4[(unsigned)j0 >> 2];   // 4 z_j values, one b128 load

        v4f r0, r1, r2, r3, r4, r5, r6, r7;      // r<h>[k]
#pragma unroll
        for (int k = 0; k < 4; ++k) {
            float fx = rintf(fmaf(-(cjf + (float)k), K, xi16));
            fx = fminf(fmaxf(fx, -16.0f), 16.0f);
            const int ix = (int)fx + 16;         // [0, 33)

            float fz = rintf(fmaf(zj4[k], -16.0f, zi16));
            fz = fminf(fmaxf(fz, -16.0f), 16.0f);
            const int iz = (int)fz + 82;         // [66, 99)

            const v4f xa = t4[2 * ix], xb = t4[2 * ix + 1];
            const v4f za = t4[2 * iz], zb = t4[2 * iz + 1];

            r0[k] = xa.x + ya.x + za.x;
            r1[k] = xa.y + ya.y + za.y;
            r2[k] = xa.z + ya.z + za.z;
            r3[k] = xa.w + ya.w + za.w;
            r4[k] = xb.x + yb.x + zb.x;
            r5[k] = xb.y + yb.y + zb.y;
            r6[k] = xb.z + yb.z + zb.z;
            r7[k] = xb.w + yb.w + zb.w;
        }

        // 8 coalesced non-temporal b128 stores (512B contiguous per wave-store)
        __builtin_nontemporal_store(r0, (v4f*)(oh0 + j0));
        __builtin_nontemporal_store(r1, (v4f*)(oh1 + j0));
        __builtin_nontemporal_store(r2, (v4f*)(oh2 + j0));
        __builtin_nontemporal_store(r3, (v4f*)(oh3 + j0));
        __builtin_nontemporal_store(r4, (v4f*)(oh4 + j0));
        __builtin_nontemporal_store(r5, (v4f*)(oh5 + j0));
        __builtin_nontemporal_store(r6, (v4f*)(oh6 + j0));
        __builtin_nontemporal_store(r7, (v4f*)(oh7 + j0));
    }
}

// ---------------------------------------------------------------------------
extern "C" void kernel_launch(void* const* d_in, const int* in_sizes, int n_in,
                              void* d_out, int out_size, void* d_ws, size_t ws_size,
                              hipStream_t stream)
{
    const float* depth = (const float*)d_in[0];   // (2,48,48)
    const float* table = (const float*)d_in[1];   // (99,8)
    float* out = (float*)d_out;                   // (2,8,2304,2304)
    float* zws = (float*)d_ws;                    // 2*2304 floats scratch

    rpe_zprep_kernel<<<dim3(2), dim3(256), 0, stream>>>(depth, zws);
    rpe_main_kernel<<<dim3(NPTS, 2), dim3(192), 0, stream>>>(table, zws, out);
}